// PointNetSetAbstractionMsg_3143916061238
// MI455X (gfx1250) — compile-verified
//
#include <hip/hip_runtime.h>
#include <cstddef>
#include <cstdint>

#define BATCH 8
#define NPTS  4096
#define NSAMP 1024

typedef __attribute__((ext_vector_type(2))) float v2f;
typedef __attribute__((ext_vector_type(8))) float v8f;

// ---------------------------------------------------------------------------
// FPS: one 1024-thread block per batch. xyz + running min-dist live in LDS.
// Emits new_xyz both as [B,S,3] scratch and directly as d_out's [B,3,S] slab.
// ---------------------------------------------------------------------------
__global__ __launch_bounds__(1024) void fps_kernel(const float* __restrict__ xyz,
                                                   float* __restrict__ new_xyz_ws,
                                                   float* __restrict__ out_xyz)
{
    __shared__ float sx[NPTS], sy[NPTS], sz[NPTS], sd[NPTS];
    __shared__ float bv[1024];
    __shared__ int   bi[1024];
    const int b = blockIdx.x;
    const int t = threadIdx.x;
    const float* X = xyz + (size_t)b * 3 * NPTS;
    for (int i = t; i < NPTS; i += 1024) {
        sx[i] = X[i]; sy[i] = X[NPTS + i]; sz[i] = X[2 * NPTS + i];
        sd[i] = 3.4e38f;
    }
    __syncthreads();
    int far = 0;
    for (int s = 0; s < NSAMP; ++s) {
        const float fx = sx[far], fy = sy[far], fz = sz[far];
        if (t == 0) {
            new_xyz_ws[((size_t)b * NSAMP + s) * 3 + 0] = fx;
            new_xyz_ws[((size_t)b * NSAMP + s) * 3 + 1] = fy;
            new_xyz_ws[((size_t)b * NSAMP + s) * 3 + 2] = fz;
            out_xyz[(size_t)b * 3 * NSAMP + 0 * NSAMP + s] = fx;
            out_xyz[(size_t)b * 3 * NSAMP + 1 * NSAMP + s] = fy;
            out_xyz[(size_t)b * 3 * NSAMP + 2 * NSAMP + s] = fz;
        }
        float bestv = -1.0f; int besti = 0;
        for (int i = t; i < NPTS; i += 1024) {
            const float dx = sx[i] - fx, dy = sy[i] - fy, dz = sz[i] - fz;
            const float d  = dx * dx + dy * dy + dz * dz;
            float nd = sd[i]; nd = nd < d ? nd : d; sd[i] = nd;
            if (nd > bestv) { bestv = nd; besti = i; }
        }
        bv[t] = bestv; bi[t] = besti;
        __syncthreads();
        for (int off = 512; off > 0; off >>= 1) {
            if (t < off && bv[t + off] > bv[t]) { bv[t] = bv[t + off]; bi[t] = bi[t + off]; }
            __syncthreads();
        }
        far = bi[0];
        __syncthreads();
    }
}

// ---------------------------------------------------------------------------
// Ball query: one thread per center; batch xyz cached in LDS (48 KB).
// Ascending-n scan taking first K within radius == reference's sort+slice.
// ---------------------------------------------------------------------------
__global__ __launch_bounds__(256) void ballquery_kernel(const float* __restrict__ xyz,
                                                        const float* __restrict__ new_xyz_ws,
                                                        int* __restrict__ idx,
                                                        float r2, int K)
{
    __shared__ float sx[NPTS], sy[NPTS], sz[NPTS];
    const int b = blockIdx.y;
    const int s = blockIdx.x * 256 + threadIdx.x;
    const float* X = xyz + (size_t)b * 3 * NPTS;
    for (int i = threadIdx.x; i < NPTS; i += 256) {
        sx[i] = X[i]; sy[i] = X[NPTS + i]; sz[i] = X[2 * NPTS + i];
    }
    __syncthreads();
    const size_t bs = (size_t)b * NSAMP + s;
    const float cx = new_xyz_ws[bs * 3 + 0];
    const float cy = new_xyz_ws[bs * 3 + 1];
    const float cz = new_xyz_ws[bs * 3 + 2];
    int* o = idx + bs * K;
    int cnt = 0, first = 0;
    for (int n = 0; n < NPTS && cnt < K; ++n) {
        const float dx = sx[n] - cx, dy = sy[n] - cy, dz = sz[n] - cz;
        if (dx * dx + dy * dy + dz * dz <= r2) {
            if (cnt == 0) first = n;
            o[cnt++] = n;
        }
    }
    if (cnt == 0) { o[0] = 0; cnt = 1; first = 0; }   // cannot happen (center in set), safety
    for (int k = cnt; k < K; ++k) o[k] = first;
}

// ---------------------------------------------------------------------------
// Group: build h0 rows [pts(6) | rel_xyz(3) | zero pad(3)] -> stride 12.
// ---------------------------------------------------------------------------
__global__ __launch_bounds__(256) void group_kernel(const float* __restrict__ xyz,
                                                    const float* __restrict__ pts,
                                                    const float* __restrict__ new_xyz_ws,
                                                    const int* __restrict__ idx,
                                                    float* __restrict__ h0, int K)
{
    const size_t tid   = (size_t)blockIdx.x * 256 + threadIdx.x;
    const size_t total = (size_t)BATCH * NSAMP * K;
    if (tid >= total) return;
    const size_t bs = tid / K;
    const int b = (int)(bs / NSAMP);
    const int n = idx[tid];
    const float* X = xyz + (size_t)b * 3 * NPTS;
    const float* P = pts + (size_t)b * 6 * NPTS;
    float* row = h0 + tid * 12;
#pragma unroll
    for (int c = 0; c < 6; ++c) row[c] = P[(size_t)c * NPTS + n];
    row[6] = X[n]            - new_xyz_ws[bs * 3 + 0];
    row[7] = X[NPTS + n]     - new_xyz_ws[bs * 3 + 1];
    row[8] = X[2 * NPTS + n] - new_xyz_ws[bs * 3 + 2];
    row[9] = 0.f; row[10] = 0.f; row[11] = 0.f;
}

// ---------------------------------------------------------------------------
// Pack weights [Cout,Cin] -> zero-padded [Cout,CinPad]; also zero stats.
// ---------------------------------------------------------------------------
__global__ __launch_bounds__(256) void packw_kernel(const float* __restrict__ W,
                                                    float* __restrict__ Wp,
                                                    int Cin, int CinPad, int Cout,
                                                    float* __restrict__ ssum,
                                                    float* __restrict__ ssq)
{
    const int i = blockIdx.x * 256 + threadIdx.x;
    if (i < Cout) { ssum[i] = 0.f; ssq[i] = 0.f; }
    if (i < Cout * CinPad) {
        const int o = i / CinPad, k = i % CinPad;
        Wp[i] = (k < Cin) ? W[o * Cin + k] : 0.f;
    }
}

// ---------------------------------------------------------------------------
// 1x1-conv as WMMA GEMM: D(16x16) += A(16x4) x B(4x16) over CinPad.
// Wave = one 16-row x 16-channel tile; block = 4 waves on one channel tile.
// Previous layer's BN+ReLU fused into the A-load; bias + per-channel
// sum/sumsq (for this layer's BN) produced on the way out.
// ---------------------------------------------------------------------------
__global__ __launch_bounds__(128) void conv_wmma_kernel(const float* __restrict__ A,
                                                        const float* __restrict__ Wp,
                                                        const float* __restrict__ bias,
                                                        const float* __restrict__ preScale,
                                                        const float* __restrict__ preShift,
                                                        int hasPre,
                                                        float* __restrict__ Out,
                                                        float* __restrict__ ssum,
                                                        float* __restrict__ ssq,
                                                        int rows, int CinPad, int Cout)
{
    __shared__ float sScale[128], sShift[128];
    __shared__ float lsum[16], lsq[16];
    const int t = threadIdx.x;
    if (hasPre) {
        for (int i = t; i < CinPad; i += 128) { sScale[i] = preScale[i]; sShift[i] = preShift[i]; }
    }
    if (t < 16) { lsum[t] = 0.f; lsq[t] = 0.f; }
    __syncthreads();

    const int wave = t >> 5, lane = t & 31;
    const int half = lane >> 4, l15 = lane & 15;
    const int rowTile = blockIdx.x * 4 + wave;
    const int row = rowTile * 16 + l15;           // A lane row (M)
    const int n0  = blockIdx.y * 16;
    const int n   = n0 + l15;                     // B/D lane column (N)

    const float* arow = A  + (size_t)row * CinPad + half * 2;
    const float* brow = Wp + (size_t)n   * CinPad + half * 2;

    v8f acc = {};
    for (int k = 0; k < CinPad; k += 4) {
        if (k + 16 < CinPad) {                     // uniform condition: no divergence
            __builtin_prefetch(&arow[k + 16], 0, 0);
            __builtin_prefetch(&brow[k + 16], 0, 0);
        }
        v2f a;  a.x = arow[k];  a.y = arow[k + 1];
        if (hasPre) {
            const int c0 = k + half * 2;
            a.x = fmaxf(fmaf(a.x, sScale[c0],     sShift[c0]),     0.f);
            a.y = fmaxf(fmaf(a.y, sScale[c0 + 1], sShift[c0 + 1]), 0.f);
        }
        v2f wv; wv.x = brow[k]; wv.y = brow[k + 1];
        acc = __builtin_amdgcn_wmma_f32_16x16x4_f32(false, a, false, wv,
                                                    (short)0, acc, false, false);
    }

    const float bn = bias[n];                      // all 8 acc entries share channel n
    float ls = 0.f, lq = 0.f;
#pragma unroll
    for (int v = 0; v < 8; ++v) {
        const float h = acc[v] + bn;
        const int m = v + half * 8;                // D row within tile
        Out[(size_t)(rowTile * 16 + m) * Cout + n] = h;
        ls += h; lq += h * h;
    }
    atomicAdd(&lsum[l15], ls);
    atomicAdd(&lsq[l15], lq);
    __syncthreads();
    if (t < 16) {
        atomicAdd(&ssum[n0 + t], lsum[t]);
        atomicAdd(&ssq [n0 + t], lsq [t]);
    }
    (void)rows;
}

// ---------------------------------------------------------------------------
// Fold BN stats into per-channel (scale, shift): h' = h*scale + shift.
// ---------------------------------------------------------------------------
__global__ __launch_bounds__(128) void finalize_bn_kernel(const float* __restrict__ ssum,
                                                          const float* __restrict__ ssq,
                                                          const float* __restrict__ gm,
                                                          const float* __restrict__ bt,
                                                          float* __restrict__ scale,
                                                          float* __restrict__ shift,
                                                          int Cout, float invCnt)
{
    const int c = blockIdx.x * 128 + threadIdx.x;
    if (c >= Cout) return;
    const float mean = ssum[c] * invCnt;
    const float var  = ssq[c] * invCnt - mean * mean;
    const float inv  = rsqrtf(var + 1e-5f);
    const float sc   = gm[c] * inv;
    scale[c] = sc;
    shift[c] = bt[c] - mean * sc;
}

// ---------------------------------------------------------------------------
// Final BN+ReLU fused with max over K; write into [B,320,S] slab of d_out.
// ---------------------------------------------------------------------------
__global__ __launch_bounds__(256) void maxpool_kernel(const float* __restrict__ H,
                                                      const float* __restrict__ scale,
                                                      const float* __restrict__ shift,
                                                      float* __restrict__ out_np,
                                                      int K, int Cout, int chOff)
{
    const size_t tid   = (size_t)blockIdx.x * 256 + threadIdx.x;
    const size_t total = (size_t)BATCH * NSAMP * Cout;
    if (tid >= total) return;
    const int c = (int)(tid % Cout);
    const size_t bs = tid / Cout;
    const int s = (int)(bs % NSAMP);
    const int b = (int)(bs / NSAMP);
    const float sc = scale[c], sh = shift[c];
    const float* base = H + bs * (size_t)K * Cout + c;
    float m = 0.f;                                 // ReLU output is >= 0
    for (int k = 0; k < K; ++k) {
        const float h = fmaxf(fmaf(base[(size_t)k * Cout], sc, sh), 0.f);
        m = fmaxf(m, h);
    }
    out_np[(size_t)b * 320 * NSAMP + (size_t)(chOff + c) * NSAMP + s] = m;
}

// ---------------------------------------------------------------------------
// Orchestration
// ---------------------------------------------------------------------------
extern "C" void kernel_launch(void* const* d_in, const int* in_sizes, int n_in,
                              void* d_out, int out_size, void* d_ws, size_t ws_size,
                              hipStream_t stream)
{
    (void)in_sizes; (void)n_in; (void)out_size; (void)ws_size;
    const float* xyz = (const float*)d_in[0];
    const float* pts = (const float*)d_in[1];
    float* out      = (float*)d_out;
    float* out_xyz  = out;                               // [B,3,S]
    float* out_np   = out + (size_t)BATCH * 3 * NSAMP;   // [B,320,S]

    char* ws = (char*)d_ws;
    size_t off = 0;
    auto alloc = [&](size_t bytes) -> char* {
        char* p = ws + off;
        off += (bytes + 255) & ~(size_t)255;
        return p;
    };
    float* new_xyz_ws = (float*)alloc((size_t)BATCH * NSAMP * 3 * 4);
    int*   idxbuf     = (int*)  alloc((size_t)BATCH * NSAMP * 64 * 4);
    float* Wp         = (float*)alloc((size_t)128 * 96 * 4);
    float* ssum       = (float*)alloc(128 * 4);
    float* ssq        = (float*)alloc(128 * 4);
    float* scl[2], *shf[2];
    scl[0] = (float*)alloc(128 * 4); shf[0] = (float*)alloc(128 * 4);
    scl[1] = (float*)alloc(128 * 4); shf[1] = (float*)alloc(128 * 4);
    float* bufA = (float*)alloc((size_t)BATCH * NSAMP * 64 * 96 * 4);   // <= 96 cols
    float* bufB = (float*)alloc((size_t)BATCH * NSAMP * 64 * 128 * 4);  // <= 128 cols

    fps_kernel<<<dim3(BATCH), dim3(1024), 0, stream>>>(xyz, new_xyz_ws, out_xyz);

    static const float R2[3]      = {0.01f, 0.04f, 0.16f};
    static const int   KK[3]      = {16, 32, 64};
    static const int   COUT[3][3] = {{32, 32, 64}, {64, 64, 128}, {64, 96, 128}};
    static const int   WBASE[3]   = {2, 14, 26};     // d_in index of each branch's first W
    static const int   CHOFF[3]   = {0, 64, 192};

    for (int br = 0; br < 3; ++br) {
        const int K = KK[br];
        const size_t rows = (size_t)BATCH * NSAMP * K;   // multiple of 64

        ballquery_kernel<<<dim3(NSAMP / 256, BATCH), dim3(256), 0, stream>>>(
            xyz, new_xyz_ws, idxbuf, R2[br], K);
        group_kernel<<<dim3((unsigned)(rows / 256)), dim3(256), 0, stream>>>(
            xyz, pts, new_xyz_ws, idxbuf, bufA, K);

        int cin = 12, cinAct = 9;
        float* inBuf = bufA;
        float* outBuf = bufB;
        for (int l = 0; l < 3; ++l) {
            const int cout = COUT[br][l];
            const float* W  = (const float*)d_in[WBASE[br] + 4 * l + 0];
            const float* bb = (const float*)d_in[WBASE[br] + 4 * l + 1];
            const float* gm = (const float*)d_in[WBASE[br] + 4 * l + 2];
            const float* bt = (const float*)d_in[WBASE[br] + 4 * l + 3];

            packw_kernel<<<dim3((cout * cin + 255) / 256), dim3(256), 0, stream>>>(
                W, Wp, cinAct, cin, cout, ssum, ssq);

            const float* ps = (l == 0) ? nullptr : scl[(l + 1) & 1];
            const float* ph = (l == 0) ? nullptr : shf[(l + 1) & 1];
            conv_wmma_kernel<<<dim3((unsigned)(rows / 64), cout / 16), dim3(128), 0, stream>>>(
                inBuf, Wp, bb, ps, ph, (l == 0) ? 0 : 1,
                outBuf, ssum, ssq, (int)rows, cin, cout);

            finalize_bn_kernel<<<dim3(1), dim3(128), 0, stream>>>(
                ssum, ssq, gm, bt, scl[l & 1], shf[l & 1], cout, 1.0f / (float)rows);

            cin = cout; cinAct = cout;
            float* tmp = inBuf; inBuf = outBuf; outBuf = tmp;
        }
        // After 3 swaps the final activations live in inBuf; layer-2 BN in slot 0.
        const int coutF = COUT[br][2];
        const size_t nmax = (size_t)BATCH * NSAMP * coutF;
        maxpool_kernel<<<dim3((unsigned)((nmax + 255) / 256)), dim3(256), 0, stream>>>(
            inBuf, scl[0], shf[0], out_np, K, coutF, CHOFF[br]);
    }
}